// DecoderAttn_5729486372924
// MI455X (gfx1250) — compile-verified
//
#include <hip/hip_runtime.h>
#include <math.h>

// MI455X (gfx1250) plan:
//  - Workload: chain of batch-1 GEMVs, 2.57 GB fp32 weight traffic/call,
//    ~0.64 GFLOP -> HBM-bound (roofline ~110us @ 23.3 TB/s), dominated by the
//    2.1 GB W_out GEMV -> coalesced wave32 streaming GEMV kernels.
//  - attn_weights @ encoder_outputs runs on V_WMMA_F32_16X16X4_F32 with
//    coalesced row-major B tiles; 4x unrolled + dual accumulators + L split
//    over gridDim.y=4 so loads batch ahead of the WMMAs (one loadcnt wait per
//    16 rows instead of per 4).
//  - Softmax: deterministic online (max,sumexp) block stats + combine
//    (no float atomics -> bit-stable across graph replays).

typedef __attribute__((ext_vector_type(2))) float v2f;
typedef __attribute__((ext_vector_type(8))) float v8f;

__device__ __forceinline__ float wave_reduce_sum(float v) {
  #pragma unroll
  for (int off = 16; off > 0; off >>= 1)
    v += __shfl_down(v, off, 32);
  return v;
}

// out[r] = dot(W[r,:], concat(p[0:n1], q[0:n2])) + bias[r] (+ out[r] if accum)
// one wave per row; float4 streaming loads, 2x unrolled (cols % 256 == 0).
__global__ void gemv_rows(const float* __restrict__ W,
                          const float* __restrict__ p, int n1,
                          const float* __restrict__ q, int n2,
                          const float* __restrict__ bias,
                          float* __restrict__ out, int rows, int accum) {
  const int wave = threadIdx.x >> 5;
  const int lane = threadIdx.x & 31;
  const int r = blockIdx.x * 8 + wave;
  if (r >= rows) return;
  const int cols = n1 + n2;
  const float4* __restrict__ Wr =
      reinterpret_cast<const float4*>(W + (size_t)r * (size_t)cols);
  const float4* __restrict__ p4 = reinterpret_cast<const float4*>(p);
  const float4* __restrict__ q4 = reinterpret_cast<const float4*>(q);
  const int nq = cols >> 2;   // multiple of 64 for all call sites
  const int n1q = n1 >> 2;
  float acc0 = 0.f, acc1 = 0.f;
  for (int i = lane; i < nq; i += 64) {
    const int j = i + 32;
    float4 w0 = Wr[i];
    float4 w1 = Wr[j];
    float4 u0 = (i < n1q) ? p4[i] : q4[i - n1q];
    float4 u1 = (j < n1q) ? p4[j] : q4[j - n1q];
    acc0 += w0.x * u0.x + w0.y * u0.y + w0.z * u0.z + w0.w * u0.w;
    acc1 += w1.x * u1.x + w1.y * u1.y + w1.z * u1.z + w1.w * u1.w;
  }
  float acc = wave_reduce_sum(acc0 + acc1);
  if (lane == 0) {
    float res = acc + (bias ? bias[r] : 0.f);
    if (accum) res += out[r];
    out[r] = res;
  }
}

__device__ __forceinline__ void sm_combine(volatile float* sm, volatile float* ss,
                                           int a, int b) {
  float m1 = sm[a], s1 = ss[a], m2 = sm[b], s2 = ss[b];
  if (s2 > 0.f) {
    if (!(s1 > 0.f)) { sm[a] = m2; ss[a] = s2; }
    else {
      float mm = fmaxf(m1, m2);
      ss[a] = s1 * expf(m1 - mm) + s2 * expf(m2 - mm);
      sm[a] = mm;
    }
  }
}

// per-block online softmax stats: stats[2b]=max, stats[2b+1]=sum exp(x-max)
__global__ void softmax_stats(const float* __restrict__ x, int n,
                              float* __restrict__ stats) {
  __shared__ float sm[256], ss[256];
  float m = -INFINITY, s = 0.f;
  for (int i = blockIdx.x * blockDim.x + threadIdx.x; i < n;
       i += gridDim.x * blockDim.x) {
    float v = x[i];
    if (v > m) { s = s * expf(m - v) + 1.f; m = v; }
    else       { s += expf(v - m); }
  }
  sm[threadIdx.x] = m; ss[threadIdx.x] = s;
  __syncthreads();
  for (int stride = 128; stride > 0; stride >>= 1) {
    if (threadIdx.x < stride) sm_combine(sm, ss, threadIdx.x, threadIdx.x + stride);
    __syncthreads();
  }
  if (threadIdx.x == 0) { stats[2 * blockIdx.x] = sm[0]; stats[2 * blockIdx.x + 1] = ss[0]; }
}

// combine <=256 block stats -> Z = max + log(sumexp)  (log_softmax offset)
__global__ void softmax_combine(const float* __restrict__ stats, int nb,
                                float* __restrict__ Z) {
  __shared__ float sm[256], ss[256];
  float m = -INFINITY, s = 0.f;
  if (threadIdx.x < nb) { m = stats[2 * threadIdx.x]; s = stats[2 * threadIdx.x + 1]; }
  sm[threadIdx.x] = m; ss[threadIdx.x] = s;
  __syncthreads();
  for (int stride = 128; stride > 0; stride >>= 1) {
    if (threadIdx.x < stride) sm_combine(sm, ss, threadIdx.x, threadIdx.x + stride);
    __syncthreads();
  }
  if (threadIdx.x == 0) *Z = sm[0] + logf(ss[0]);
}

// partial attn_applied over an L/4 slice via V_WMMA_F32_16X16X4_F32.
// Block (bx,by): columns [16bx,16bx+16), rows [by*L/4, (by+1)*L/4).
// A layout (ISA 7.12.2, 32-bit 16x4): lanes0-15 vgpr{0,1}=K{0,1}, lanes16-31=K{2,3}.
// B mirrored (matches SWMMAC B tables): same (half,vgpr)->K map, N=lane&15.
// D (documented): lane n<16, vgpr0 = D[0][n]; A rows are broadcast so every D
// row equals the per-column partial dot product.
__global__ void wmma_attn_apply(const float* __restrict__ E,      // L x H
                                const float* __restrict__ logits, // L
                                const float* __restrict__ Zp,
                                float* __restrict__ part,         // 4 x H
                                int L, int H) {
  __shared__ float red[8][16];
  const float Z = *Zp;
  const int wave = threadIdx.x >> 5;
  const int lane = threadIdx.x & 31;
  const int half = lane >> 4;
  const int ln   = lane & 15;
  const int col  = blockIdx.x * 16 + ln;
  const int blkChunk  = L >> 2;              // rows per gridDim.y slice
  const int waveChunk = blkChunk >> 3;       // rows per wave (L/32 = 128)
  const int l0beg = blockIdx.y * blkChunk + wave * waveChunk;
  v8f c0 = {}, c1 = {};
  for (int l0 = l0beg; l0 < l0beg + waveChunk; l0 += 16) {
    v2f a[4], b[4];
    #pragma unroll
    for (int u = 0; u < 4; ++u) {           // batch all 16 loads up front
      const int row = l0 + 4 * u + 2 * half;
      a[u].x = logits[row]     - Z;          // K = 0/2
      a[u].y = logits[row + 1] - Z;          // K = 1/3
      b[u].x = E[(size_t)row * H + col];     // 64B coalesced per half-wave
      b[u].y = E[(size_t)(row + 1) * H + col];
    }
    c0 = __builtin_amdgcn_wmma_f32_16x16x4_f32(false, a[0], false, b[0], (short)0, c0, false, false);
    c1 = __builtin_amdgcn_wmma_f32_16x16x4_f32(false, a[1], false, b[1], (short)0, c1, false, false);
    c0 = __builtin_amdgcn_wmma_f32_16x16x4_f32(false, a[2], false, b[2], (short)0, c0, false, false);
    c1 = __builtin_amdgcn_wmma_f32_16x16x4_f32(false, a[3], false, b[3], (short)0, c1, false, false);
  }
  if (lane < 16) red[wave][ln] = c0[0] + c1[0];   // D[0][n]
  __syncthreads();
  if (threadIdx.x < 16) {
    float s = 0.f;
    #pragma unroll
    for (int w = 0; w < 8; ++w) s += red[w][threadIdx.x];
    part[(size_t)blockIdx.y * H + blockIdx.x * 16 + threadIdx.x] = s;
  }
}

// attn_applied[j] = sum of the 4 L-slice partials (deterministic reduce)
__global__ void reduce_part4(const float* __restrict__ part,
                             float* __restrict__ out, int H) {
  int j = blockIdx.x * blockDim.x + threadIdx.x;
  if (j < H) out[j] = (part[j] + part[H + j]) + (part[2 * H + j] + part[3 * H + j]);
}

// gates (i,f,g,o) -> h_new  (c_new is consumed here, never materialized)
__global__ void lstm_elem(const float* __restrict__ gates,
                          const float* __restrict__ c0,
                          float* __restrict__ hnew, int Hn) {
  int i = blockIdx.x * blockDim.x + threadIdx.x;
  if (i >= Hn) return;
  float ig = gates[i], fg = gates[Hn + i], gg = gates[2 * Hn + i], og = gates[3 * Hn + i];
  float si = 1.f / (1.f + expf(-ig));
  float sf = 1.f / (1.f + expf(-fg));
  float so = 1.f / (1.f + expf(-og));
  float cn = sf * c0[i] + si * tanhf(gg);
  hnew[i] = so * tanhf(cn);
}

__global__ void sub_scalar(float* __restrict__ x, const float* __restrict__ Zp, int n) {
  int i = blockIdx.x * blockDim.x + threadIdx.x;
  if (i < n) x[i] = x[i] - *Zp;
}

extern "C" void kernel_launch(void* const* d_in, const int* in_sizes, int n_in,
                              void* d_out, int out_size, void* d_ws, size_t ws_size,
                              hipStream_t stream) {
  const int H = 4096, L = 4096, V = 128000;
  const float* E      = (const float*)d_in[0];
  const float* h0     = (const float*)d_in[1];
  const float* c0     = (const float*)d_in[2];
  const float* x0     = (const float*)d_in[3];
  const float* W_attn = (const float*)d_in[4];
  const float* b_attn = (const float*)d_in[5];
  const float* W_comb = (const float*)d_in[6];
  const float* b_comb = (const float*)d_in[7];
  const float* W_ih   = (const float*)d_in[8];
  const float* b_ih   = (const float*)d_in[9];
  const float* W_hh   = (const float*)d_in[10];
  const float* b_hh   = (const float*)d_in[11];
  const float* W_out  = (const float*)d_in[12];
  const float* b_out  = (const float*)d_in[13];
  float* out = (float*)d_out;
  float* ws  = (float*)d_ws;

  float* logits    = ws;           // 4096
  float* attn_app  = ws + 4096;    // 4096
  float* lstm_in   = ws + 8192;    // 4096
  float* gates     = ws + 12288;   // 16384
  float* hnew      = ws + 28672;   // 4096
  float* stats     = ws + 32768;   // 512
  float* Z1        = ws + 33280;   // 1
  float* Z2        = ws + 33281;   // 1
  float* attn_part = ws + 33792;   // 4 * 4096 = 16384

  dim3 b256(256);

  // 1) attn logits = W_attn @ [h;x] + b_attn         (134 MB stream)
  gemv_rows<<<L / 8, b256, 0, stream>>>(W_attn, h0, H, x0, H, b_attn, logits, L, 0);
  // 2) log_softmax offset Z1 over L
  softmax_stats<<<8, b256, 0, stream>>>(logits, L, stats);
  softmax_combine<<<1, b256, 0, stream>>>(stats, 8, Z1);
  // 3) attn_applied = (logits - Z1) @ E              (67 MB, WMMA path, 1024 blocks)
  wmma_attn_apply<<<dim3(H / 16, 4), b256, 0, stream>>>(E, logits, Z1, attn_part, L, H);
  reduce_part4<<<H / 256, b256, 0, stream>>>(attn_part, attn_app, H);
  // 4) lstm_in = W_comb @ [x; attn_applied] + b_comb (134 MB)
  gemv_rows<<<H / 8, b256, 0, stream>>>(W_comb, x0, H, attn_app, H, b_comb, lstm_in, H, 0);
  // 5) gates = W_ih @ lstm_in + b_ih + W_hh @ h + b_hh (268 MB)
  gemv_rows<<<(4 * H) / 8, b256, 0, stream>>>(W_ih, lstm_in, H, lstm_in, 0, b_ih, gates, 4 * H, 0);
  gemv_rows<<<(4 * H) / 8, b256, 0, stream>>>(W_hh, h0, H, h0, 0, b_hh, gates, 4 * H, 1);
  // 6) LSTM elementwise -> h_new
  lstm_elem<<<H / 256, b256, 0, stream>>>(gates, c0, hnew, H);
  // 7) word logits = W_out @ h_new + b_out           (2.10 GB, dominant)
  gemv_rows<<<V / 8, b256, 0, stream>>>(W_out, hnew, H, hnew, 0, b_out, out, V, 0);
  // 8) log_softmax over V, in place
  softmax_stats<<<256, b256, 0, stream>>>(out, V, stats);
  softmax_combine<<<1, b256, 0, stream>>>(stats, 256, Z2);
  sub_scalar<<<(V + 255) / 256, b256, 0, stream>>>(out, Z2, V);
}